// PositionFilter_51539607552084
// MI455X (gfx1250) — compile-verified
//
#include <hip/hip_runtime.h>
#include <hip/hip_bf16.h>
#include <stdint.h>

// ---------------------------------------------------------------------------
// Batched EKF predict+update, one tracklet per thread.
// CDNA5 path: TDM tensor_load_to_lds stages each block's P (and Q) tile into
// LDS with descriptor-side padding (68-float slots) for low-conflict ds reads.
// Small per-thread loads + x_pred math overlap the in-flight DMA; all
// streaming global traffic carries non-temporal hints (no L2 reuse).
// ---------------------------------------------------------------------------

#define EKF_THREADS 64
#define EKF_SLOT    68            // 64 floats + 4 pad floats per tracklet slot
#define EKF_LDS_F   (EKF_THREADS * EKF_SLOT)

#ifndef __has_builtin
#define __has_builtin(x) 0
#endif

#if __has_builtin(__builtin_amdgcn_tensor_load_to_lds)
#define EKF_HAVE_TDM 1
#else
#define EKF_HAVE_TDM 0
#endif

typedef __attribute__((ext_vector_type(4))) unsigned int v4u_t;
typedef __attribute__((ext_vector_type(8))) int          v8i_t;
typedef __attribute__((ext_vector_type(4))) int          v4i_t;
typedef __attribute__((ext_vector_type(4))) float        v4f_t;  // clang vector: OK for NT builtins

#if EKF_HAVE_TDM
// Issue one 1-D TDM copy of 16 KB (64 tracklets x 64 f32) into LDS, inserting
// 4 dwords of padding after every 64 dwords (=> 68-float slots in LDS).
__device__ __forceinline__ void ekf_tdm_load(const float* gsrc, uint32_t lds_byte_off) {
  const uint64_t ga = (uint64_t)(uintptr_t)gsrc;
  // D# group 0: count=1 | lds_addr | global_addr[56:0] | type=2
  v4u_t g0 = { 1u,
               lds_byte_off,
               (uint32_t)ga,
               (uint32_t)((ga >> 32) & 0x01FFFFFFu) | 0x80000000u };
  const uint32_t dim0 = 2048u;  // 16384 bytes in 8-byte units
  // D# group 1:
  //  w0: data_size=3 (8B) | pad_enable | pad_interval=5 (64 dw) | pad_amount=3 (4 dw)
  //  w1: tensor_dim0[15:0] << 16
  //  w2: tensor_dim0[31:16] | tensor_dim1(=1) << 16
  //  w3: tile_dim0 << 16
  //  w5: tensor_dim0_stride
  v8i_t g1 = { (int)((3u << 16) | (1u << 20) | (5u << 22) | (3u << 25)),
               (int)((dim0 & 0xFFFFu) << 16),
               (int)(((dim0 >> 16) & 0xFFFFu) | (1u << 16)),
               (int)((dim0 & 0xFFFFu) << 16),
               0,
               (int)dim0,
               0, 0 };
  v4i_t gz4 = { 0, 0, 0, 0 };
  v8i_t gz8 = { 0, 0, 0, 0, 0, 0, 0, 0 };
  // clang-23 / amdgpu-toolchain 6-arg form:
  // (g0, g1, g2, g3, extra, cpol) -- unused groups zero-filled (1-D copy).
  __builtin_amdgcn_tensor_load_to_lds(g0, g1, gz4, gz4, gz8, 0);
}
#endif

// Fallback: cooperative coalesced staging into the same padded LDS layout.
__device__ __forceinline__ void ekf_coop_stage(const float* __restrict__ gsrc,
                                               float* lds) {
  const v4f_t* g4 = (const v4f_t*)gsrc;
  v4f_t* l4 = (v4f_t*)lds;
  const int t = (int)threadIdx.x;
#pragma unroll
  for (int k = 0; k < 16; ++k) {
    int v = t + EKF_THREADS * k;      // float4 index within block tile [0,1024)
    int trk = v >> 4, e = v & 15;     // tracklet slot, element-quad
    l4[trk * 17 + e] = g4[v];
  }
}

__global__ __launch_bounds__(EKF_THREADS)
void ekf_predict_update_kernel(const float* __restrict__ x,
                               const float* __restrict__ P,
                               const float* __restrict__ z,
                               const float* __restrict__ Q,
                               const float* __restrict__ R,
                               float* __restrict__ xo,
                               float* __restrict__ Po,
                               int n, int qFull, int rFull) {
  __shared__ float ldsP[EKF_LDS_F];
  __shared__ float ldsQ[EKF_LDS_F];

  const int t    = (int)threadIdx.x;
  const int blk0 = (int)blockIdx.x * EKF_THREADS;
  const int b    = blk0 + t;
  const bool fullBlock = (blk0 + EKF_THREADS) <= n;
  const bool active    = b < n;

  const float dt  = 0.1f;
  const float dt2 = 0.005f;  // 0.5*dt*dt

  // ---- kick off the block DMA first ------------------------------------
  if (fullBlock) {
#if EKF_HAVE_TDM
    if (t == 0) {
      ekf_tdm_load(P + (size_t)blk0 * 64, (uint32_t)(uintptr_t)&ldsP[0]);
      if (qFull) ekf_tdm_load(Q + (size_t)blk0 * 64, (uint32_t)(uintptr_t)&ldsQ[0]);
    }
#else
    ekf_coop_stage(P + (size_t)blk0 * 64, ldsP);
    if (qFull) ekf_coop_stage(Q + (size_t)blk0 * 64, ldsQ);
#endif
  }

  // ---- overlap: small per-thread loads + x_pred while DMA is in flight --
  float xs[8];
  float zv0 = 0.f, zv1 = 0.f, zv2 = 0.f;
  float Rm[9];
  if (active) {
    __builtin_prefetch(x + (size_t)b * 8, 0, 0);
    __builtin_prefetch(z + (size_t)b * 3, 0, 0);
    {
      const v4f_t* g = (const v4f_t*)(x + (size_t)b * 8);
      v4f_t a = __builtin_nontemporal_load(g + 0);
      v4f_t c = __builtin_nontemporal_load(g + 1);
      xs[0] = a.x; xs[1] = a.y; xs[2] = a.z; xs[3] = a.w;
      xs[4] = c.x; xs[5] = c.y; xs[6] = c.z; xs[7] = c.w;
    }
    // x_pred = F x (sparse F)
    xs[0] += dt * xs[3] + dt2 * xs[6];
    xs[1] += dt * xs[4] + dt2 * xs[7];
    xs[2] += dt * xs[5];
    xs[3] += dt * xs[6];
    xs[4] += dt * xs[7];

    const float* zp = z + (size_t)b * 3;
    zv0 = __builtin_nontemporal_load(zp + 0);
    zv1 = __builtin_nontemporal_load(zp + 1);
    zv2 = __builtin_nontemporal_load(zp + 2);

    if (rFull) {
      const float* rp = R + (size_t)b * 9;
#pragma unroll
      for (int k = 0; k < 9; ++k) Rm[k] = __builtin_nontemporal_load(rp + k);
    } else {
#pragma unroll
      for (int k = 0; k < 9; ++k) Rm[k] = R[k];   // uniform -> scalar loads
    }
  }

  // ---- wait for DMA, publish LDS to the workgroup -----------------------
  if (fullBlock) {
#if EKF_HAVE_TDM
    __builtin_amdgcn_s_wait_tensorcnt(0);
#endif
    __syncthreads();
  }

  if (!active) return;

  // ---- load P into registers -------------------------------------------
  float p[8][8];
  if (fullBlock) {
    const v4f_t* s = (const v4f_t*)&ldsP[(size_t)t * EKF_SLOT];
#pragma unroll
    for (int i = 0; i < 8; ++i) {
      v4f_t a = s[2 * i + 0], c = s[2 * i + 1];
      p[i][0] = a.x; p[i][1] = a.y; p[i][2] = a.z; p[i][3] = a.w;
      p[i][4] = c.x; p[i][5] = c.y; p[i][6] = c.z; p[i][7] = c.w;
    }
  } else {
    const v4f_t* g = (const v4f_t*)(P + (size_t)b * 64);
#pragma unroll
    for (int i = 0; i < 8; ++i) {
      v4f_t a = __builtin_nontemporal_load(g + 2 * i + 0);
      v4f_t c = __builtin_nontemporal_load(g + 2 * i + 1);
      p[i][0] = a.x; p[i][1] = a.y; p[i][2] = a.z; p[i][3] = a.w;
      p[i][4] = c.x; p[i][5] = c.y; p[i][6] = c.z; p[i][7] = c.w;
    }
  }

  // ---- P_pred = F P F^T + Q, exploiting sparse F ------------------------
  // Row pass: A = F P   (rows 0..4 change; reads use pre-update rows 3..7)
#pragma unroll
  for (int j = 0; j < 8; ++j) {
    p[0][j] += dt * p[3][j] + dt2 * p[6][j];
    p[1][j] += dt * p[4][j] + dt2 * p[7][j];
    p[2][j] += dt * p[5][j];
    p[3][j] += dt * p[6][j];
    p[4][j] += dt * p[7][j];
  }
  // Column pass: P_pred = A F^T
#pragma unroll
  for (int i = 0; i < 8; ++i) {
    p[i][0] += dt * p[i][3] + dt2 * p[i][6];
    p[i][1] += dt * p[i][4] + dt2 * p[i][7];
    p[i][2] += dt * p[i][5];
    p[i][3] += dt * p[i][6];
    p[i][4] += dt * p[i][7];
  }
  // + Q
  if (qFull) {
    if (fullBlock) {
      const v4f_t* s = (const v4f_t*)&ldsQ[(size_t)t * EKF_SLOT];
#pragma unroll
      for (int i = 0; i < 8; ++i) {
        v4f_t a = s[2 * i + 0], c = s[2 * i + 1];
        p[i][0] += a.x; p[i][1] += a.y; p[i][2] += a.z; p[i][3] += a.w;
        p[i][4] += c.x; p[i][5] += c.y; p[i][6] += c.z; p[i][7] += c.w;
      }
    } else {
      const float* q = Q + (size_t)b * 64;
#pragma unroll
      for (int i = 0; i < 8; ++i)
#pragma unroll
        for (int j = 0; j < 8; ++j) p[i][j] += __builtin_nontemporal_load(q + i * 8 + j);
    }
  } else {
#pragma unroll
    for (int i = 0; i < 8; ++i)
#pragma unroll
      for (int j = 0; j < 8; ++j) p[i][j] += Q[i * 8 + j];   // uniform
  }

  // ---- S = P_pred[0:3,0:3] + R ; invert 3x3 via adjugate ---------------
  float s00 = p[0][0] + Rm[0], s01 = p[0][1] + Rm[1], s02 = p[0][2] + Rm[2];
  float s10 = p[1][0] + Rm[3], s11 = p[1][1] + Rm[4], s12 = p[1][2] + Rm[5];
  float s20 = p[2][0] + Rm[6], s21 = p[2][1] + Rm[7], s22 = p[2][2] + Rm[8];

  float c00 = s11 * s22 - s12 * s21;
  float c01 = s12 * s20 - s10 * s22;
  float c02 = s10 * s21 - s11 * s20;
  float det = s00 * c00 + s01 * c01 + s02 * c02;
  float rdet = 1.0f / det;
  float i00 = c00 * rdet, i01 = (s02 * s21 - s01 * s22) * rdet, i02 = (s01 * s12 - s02 * s11) * rdet;
  float i10 = c01 * rdet, i11 = (s00 * s22 - s02 * s20) * rdet, i12 = (s02 * s10 - s00 * s12) * rdet;
  float i20 = c02 * rdet, i21 = (s01 * s20 - s00 * s21) * rdet, i22 = (s00 * s11 - s01 * s10) * rdet;

  // ---- K = P_pred[:, 0:3] * S^-1  (8x3) --------------------------------
  float K0[8], K1[8], K2[8];
#pragma unroll
  for (int i = 0; i < 8; ++i) {
    K0[i] = p[i][0] * i00 + p[i][1] * i10 + p[i][2] * i20;
    K1[i] = p[i][0] * i01 + p[i][1] * i11 + p[i][2] * i21;
    K2[i] = p[i][0] * i02 + p[i][1] * i12 + p[i][2] * i22;
  }

  // ---- x_upd = x_pred + K (z - H x_pred) -------------------------------
  float in0 = zv0 - xs[0];
  float in1 = zv1 - xs[1];
  float in2 = zv2 - xs[2];
  {
    v4f_t o0, o1;
    o0.x = xs[0] + K0[0] * in0 + K1[0] * in1 + K2[0] * in2;
    o0.y = xs[1] + K0[1] * in0 + K1[1] * in1 + K2[1] * in2;
    o0.z = xs[2] + K0[2] * in0 + K1[2] * in1 + K2[2] * in2;
    o0.w = xs[3] + K0[3] * in0 + K1[3] * in1 + K2[3] * in2;
    o1.x = xs[4] + K0[4] * in0 + K1[4] * in1 + K2[4] * in2;
    o1.y = xs[5] + K0[5] * in0 + K1[5] * in1 + K2[5] * in2;
    o1.z = xs[6] + K0[6] * in0 + K1[6] * in1 + K2[6] * in2;
    o1.w = xs[7] + K0[7] * in0 + K1[7] * in1 + K2[7] * in2;
    v4f_t* og = (v4f_t*)(xo + (size_t)b * 8);
    __builtin_nontemporal_store(o0, og + 0);
    __builtin_nontemporal_store(o1, og + 1);
  }

  // ---- P_upd = P_pred - K (H P_pred)  (rank-3 update) ------------------
  float h0[8], h1[8], h2[8];
#pragma unroll
  for (int j = 0; j < 8; ++j) { h0[j] = p[0][j]; h1[j] = p[1][j]; h2[j] = p[2][j]; }
#pragma unroll
  for (int i = 0; i < 8; ++i)
#pragma unroll
    for (int j = 0; j < 8; ++j)
      p[i][j] -= K0[i] * h0[j] + K1[i] * h1[j] + K2[i] * h2[j];

  {
    v4f_t* og = (v4f_t*)(Po + (size_t)b * 64);
#pragma unroll
    for (int i = 0; i < 8; ++i) {
      v4f_t a, c;
      a.x = p[i][0]; a.y = p[i][1]; a.z = p[i][2]; a.w = p[i][3];
      c.x = p[i][4]; c.y = p[i][5]; c.z = p[i][6]; c.w = p[i][7];
      __builtin_nontemporal_store(a, og + 2 * i + 0);
      __builtin_nontemporal_store(c, og + 2 * i + 1);
    }
  }
}

extern "C" void kernel_launch(void* const* d_in, const int* in_sizes, int n_in,
                              void* d_out, int out_size, void* d_ws, size_t ws_size,
                              hipStream_t stream) {
  (void)n_in; (void)out_size; (void)d_ws; (void)ws_size;
  const float* x = (const float*)d_in[0];   // [B,8,1]
  const float* P = (const float*)d_in[1];   // [B,8,8]
  const float* z = (const float*)d_in[2];   // [B,3,1]
  const float* Q = (const float*)d_in[3];   // [B,8,8] or [8,8]
  const float* R = (const float*)d_in[4];   // [B,3,3] or [3,3]

  const int n = in_sizes[0] / 8;
  const int qFull = (in_sizes[3] == in_sizes[1]) ? 1 : 0;       // per-batch Q?
  const int rFull = ((long long)in_sizes[4] == 3LL * in_sizes[2]) ? 1 : 0; // per-batch R?

  float* xo = (float*)d_out;                 // [B,8,1]
  float* Po = (float*)d_out + (size_t)n * 8; // [B,8,8]

  dim3 grid((unsigned)((n + EKF_THREADS - 1) / EKF_THREADS));
  dim3 block(EKF_THREADS);
  hipLaunchKernelGGL(ekf_predict_update_kernel, grid, block, 0, stream,
                     x, P, z, Q, R, xo, Po, n, qFull, rFull);
}